// GIN_71528385348099
// MI455X (gfx1250) — compile-verified
//
#include <hip/hip_runtime.h>

typedef __attribute__((ext_vector_type(2))) float v2f;
typedef __attribute__((ext_vector_type(8))) float v8f;

#define NODES  50000
#define EDGES  800000
#define GRAPHS 512
#define PADK   4        // LDS row padding (floats) to spread rows across banks

static inline long long cdivll(long long a, long long b) { return (a + b - 1) / b; }

// ---------------------------------------------------------------------------
// Zero fill (float4-vectorized)
// ---------------------------------------------------------------------------
__global__ __launch_bounds__(256) void zero_f32(float4* __restrict__ p, long long n4) {
    long long i = blockIdx.x * (long long)blockDim.x + threadIdx.x;
    if (i < n4) p[i] = make_float4(0.f, 0.f, 0.f, 0.f);
}

// ---------------------------------------------------------------------------
// Edge scatter-add: aggr[dst] += h[src].  One thread per (edge, 4-float chunk):
// a wave reads 512B contiguous from one source row and issues
// global_atomic_add_f32 (no-return) into the destination row.
// ---------------------------------------------------------------------------
__global__ __launch_bounds__(256) void scatter_add4(
    const float* __restrict__ H, const int* __restrict__ eidx,
    float* __restrict__ aggr, int ncol4, long long total) {
    long long t = blockIdx.x * (long long)blockDim.x + threadIdx.x;
    if (t >= total) return;
    int       c4 = (int)(t % ncol4);
    long long e  = t / ncol4;
    int s = eidx[e];
    int d = eidx[EDGES + e];
    int ncol = ncol4 * 4;
    const float4 v = *(const float4*)(H + (size_t)s * ncol + (size_t)c4 * 4);
    float* o = aggr + (size_t)d * ncol + (size_t)c4 * 4;
    atomicAdd(o + 0, v.x);
    atomicAdd(o + 1, v.y);
    atomicAdd(o + 2, v.z);
    atomicAdd(o + 3, v.w);
}

// ---------------------------------------------------------------------------
// Fused GIN GEMM:  Out = act( (A [+ G]) @ W^T + bias )
//
// Block = 128 threads (4 waves).  The block owns one 16-row M-tile:
//   * cooperative stage of S = A(+G) tile into LDS (padded row stride),
//     so the h+aggr add and the HBM read of A happen ONCE per tile;
//   * each wave owns NACC 16-col N-tiles (NACC accumulators), so one LDS A
//     fragment feeds NACC back-to-back v_wmma_f32_16x16x4_f32 ops;
//   * 16-k chunks are double-buffered (loads of chunk i+1 issued before the
//     WMMAs of chunk i) to hide LDS/global latency.
//
// fp32 WMMA layouts (ISA §7.12.2, wave32):
//   A 16x4: lane -> row lane%16, K pair selected by lane/16.
//   B 4x16: lane supplies W[col][kk], W[col][kk+1], col = ntile*16+lane%16.
//   C/D   : c[r] = Out[m0 + r + 8*(lane/16)][col].
// EXEC stays all-1s through every WMMA (N-guard via zero-scaled W rows).
// ---------------------------------------------------------------------------
template <int K, int NACC, bool ADD_AGGR, bool RELU, bool GUARD_N>
__global__ __launch_bounds__(128) void wmma_gemm(
    const float* __restrict__ A, const float* __restrict__ G,
    const float* __restrict__ W, const float* __restrict__ bias,
    float* __restrict__ Out, int Nout) {
    __shared__ float As[16 * (K + PADK)];

    const int tid   = threadIdx.x;
    const int wave  = tid >> 5;
    const int lane  = tid & 31;
    const int half  = lane >> 4;
    const int l16   = lane & 15;
    const int mrow0 = blockIdx.x * 16;

    // ---- cooperative stage: As = A (+ G), row-major, padded stride ----
    const int C4 = K / 4;
    for (int i = tid; i < 16 * C4; i += 128) {
        const int r  = i / C4;
        const int c4 = i - r * C4;
        float4 v = *(const float4*)(A + (size_t)(mrow0 + r) * K + (size_t)c4 * 4);
        if constexpr (ADD_AGGR) {
            const float4 g = *(const float4*)(G + (size_t)(mrow0 + r) * K + (size_t)c4 * 4);
            v.x += g.x; v.y += g.y; v.z += g.z; v.w += g.w;
        }
        *(float4*)(&As[r * (K + PADK) + c4 * 4]) = v;
    }
    __syncthreads();

    const int ntiles  = (Nout + 15) >> 4;
    const int ngroups = (ntiles + NACC - 1) / NACC;
    if (wave >= ngroups) return;  // wave-uniform, after the only barrier

    // per-accumulator column / W row setup
    const float* wp[NACC];
    float        bv[NACC];
    int          cols[NACC];
    bool         okc[NACC];
#pragma unroll
    for (int j = 0; j < NACC; ++j) {
        const int col = (wave * NACC + j) * 16 + l16;
        cols[j] = col;
        int  wr = col;
        bool ok = true;
        if constexpr (GUARD_N) {
            if (col >= Nout) { wr = 0; ok = false; }
        }
        okc[j] = ok;
        wp[j]  = W + (size_t)wr * K;
        bv[j]  = ok ? bias[col] : 0.f;
    }

    const float* as = &As[l16 * (K + PADK)];

    auto loadChunk = [&](int k0, v2f* aa, v2f (*bb)[4]) {
#pragma unroll
        for (int u = 0; u < 4; ++u) {
            const int kk = k0 + 4 * u + 2 * half;
            aa[u].x = as[kk];
            aa[u].y = as[kk + 1];
#pragma unroll
            for (int j = 0; j < NACC; ++j) {
                float b0 = wp[j][kk];
                float b1 = wp[j][kk + 1];
                if constexpr (GUARD_N) {
                    if (!okc[j]) { b0 = 0.f; b1 = 0.f; }
                }
                bb[j][u].x = b0;
                bb[j][u].y = b1;
            }
        }
    };

    v8f acc[NACC];
#pragma unroll
    for (int j = 0; j < NACC; ++j) acc[j] = (v8f){};

    v2f a[4], b[NACC][4];
    loadChunk(0, a, b);
#pragma unroll
    for (int k0 = 0; k0 < K; k0 += 16) {
        v2f a2[4], b2[NACC][4];
        const int kn = (k0 + 16 < K) ? (k0 + 16) : 0;  // last prefetch is dummy
        loadChunk(kn, a2, b2);
#pragma unroll
        for (int u = 0; u < 4; ++u)
#pragma unroll
            for (int j = 0; j < NACC; ++j)
                acc[j] = __builtin_amdgcn_wmma_f32_16x16x4_f32(
                    false, a[u], false, b[j][u], (short)0, acc[j], false, false);
#pragma unroll
        for (int u = 0; u < 4; ++u) {
            a[u] = a2[u];
#pragma unroll
            for (int j = 0; j < NACC; ++j) b[j][u] = b2[j][u];
        }
    }

#pragma unroll
    for (int j = 0; j < NACC; ++j) {
#pragma unroll
        for (int r = 0; r < 8; ++r) {
            float v = acc[j][r] + bv[j];
            if constexpr (RELU) v = fmaxf(v, 0.f);
            const int orow = mrow0 + r + 8 * half;
            bool ok = true;
            if constexpr (GUARD_N) ok = okc[j];
            if (ok) Out[(size_t)orow * Nout + cols[j]] = v;
        }
    }
}

// ---------------------------------------------------------------------------
// Global mean pool (batch is per-node graph id)
// ---------------------------------------------------------------------------
__global__ __launch_bounds__(256) void pool_count(const int* __restrict__ batch,
                                                  float* __restrict__ counts) {
    int n = blockIdx.x * blockDim.x + threadIdx.x;
    if (n < NODES) atomicAdd(&counts[batch[n]], 1.f);
}

__global__ __launch_bounds__(256) void pool_sum4(const float* __restrict__ H,
                                                 const int* __restrict__ batch,
                                                 float* __restrict__ pooled) {
    long long t = blockIdx.x * (long long)blockDim.x + threadIdx.x;
    const long long total = (long long)NODES * 64;  // 256 cols / 4
    if (t >= total) return;
    int c4 = (int)(t % 64);
    int n  = (int)(t / 64);
    int g  = batch[n];
    const float4 v = *(const float4*)(H + (size_t)n * 256 + (size_t)c4 * 4);
    float* o = pooled + (size_t)g * 256 + (size_t)c4 * 4;
    atomicAdd(o + 0, v.x);
    atomicAdd(o + 1, v.y);
    atomicAdd(o + 2, v.z);
    atomicAdd(o + 3, v.w);
}

__global__ __launch_bounds__(256) void pool_div(float* __restrict__ pooled,
                                                const float* __restrict__ counts) {
    int i = blockIdx.x * blockDim.x + threadIdx.x;
    if (i < GRAPHS * 256) pooled[i] /= fmaxf(counts[i >> 8], 1.f);
}

// ---------------------------------------------------------------------------
extern "C" void kernel_launch(void* const* d_in, const int* in_sizes, int n_in,
                              void* d_out, int out_size, void* d_ws, size_t ws_size,
                              hipStream_t stream) {
    const float* x    = (const float*)d_in[0];
    const int*   eidx = (const int*)d_in[1];
    const int*   batc = (const int*)d_in[2];
    const float* W1 = (const float*)d_in[3];  const float* b1  = (const float*)d_in[4];
    const float* W2 = (const float*)d_in[5];  const float* b2  = (const float*)d_in[6];
    const float* W3 = (const float*)d_in[7];  const float* b3  = (const float*)d_in[8];
    const float* Wf1 = (const float*)d_in[9]; const float* bf1 = (const float*)d_in[10];
    const float* Wf2 = (const float*)d_in[11];const float* bf2 = (const float*)d_in[12];

    // workspace layout (fp32): aggr | hA | hB | pooled | counts | fc1
    float* aggr   = (float*)d_ws;
    float* hA     = aggr   + (size_t)NODES * 256;
    float* hB     = hA     + (size_t)NODES * 256;
    float* pooled = hB     + (size_t)NODES * 256;
    float* counts = pooled + (size_t)GRAPHS * 256;
    float* fc1    = counts + GRAPHS;

    auto zero = [&](float* p, long long n) {
        long long n4 = n / 4;
        zero_f32<<<(unsigned)cdivll(n4, 256), 256, 0, stream>>>((float4*)p, n4);
    };

    const dim3 blk(128);  // 4 waves / block

    // ---- GIN layer 1: K=128 -> 256 ----
    zero(aggr, (long long)NODES * 128);
    {
        long long tot = (long long)EDGES * 32;
        scatter_add4<<<(unsigned)cdivll(tot, 256), 256, 0, stream>>>(x, eidx, aggr, 32, tot);
    }
    wmma_gemm<128, 4, true, true, false>
        <<<dim3(NODES / 16), blk, 0, stream>>>(x, aggr, W1, b1, hA, 256);

    // ---- GIN layer 2: K=256 -> 256 ----
    zero(aggr, (long long)NODES * 256);
    {
        long long tot = (long long)EDGES * 64;
        scatter_add4<<<(unsigned)cdivll(tot, 256), 256, 0, stream>>>(hA, eidx, aggr, 64, tot);
    }
    wmma_gemm<256, 4, true, true, false>
        <<<dim3(NODES / 16), blk, 0, stream>>>(hA, aggr, W2, b2, hB, 256);

    // ---- GIN layer 3: K=256 -> 256 ----
    zero(aggr, (long long)NODES * 256);
    {
        long long tot = (long long)EDGES * 64;
        scatter_add4<<<(unsigned)cdivll(tot, 256), 256, 0, stream>>>(hB, eidx, aggr, 64, tot);
    }
    wmma_gemm<256, 4, true, true, false>
        <<<dim3(NODES / 16), blk, 0, stream>>>(hB, aggr, W3, b3, hA, 256);

    // ---- global mean pool ----
    zero(pooled, (long long)GRAPHS * 256 + GRAPHS);  // pooled + counts contiguous
    pool_count<<<(unsigned)cdivll(NODES, 256), 256, 0, stream>>>(batc, counts);
    {
        long long tot = (long long)NODES * 64;
        pool_sum4<<<(unsigned)cdivll(tot, 256), 256, 0, stream>>>(hA, batc, pooled);
    }
    pool_div<<<(unsigned)cdivll((long long)GRAPHS * 256, 256), 256, 0, stream>>>(pooled, counts);

    // ---- FC head ----
    wmma_gemm<256, 4, false, true, false>
        <<<dim3(GRAPHS / 16), blk, 0, stream>>>(pooled, nullptr, Wf1, bf1, fc1, 128);
    wmma_gemm<128, 1, false, false, true>
        <<<dim3(GRAPHS / 16), blk, 0, stream>>>(fc1, nullptr, Wf2, bf2, (float*)d_out, 10);
}